// LSTM_pred_85641647882936
// MI455X (gfx1250) — compile-verified
//
#include <hip/hip_runtime.h>

// 2-layer LSTM (B=8192, S=256, in=13, H=10, out=9) on gfx1250.
// One wave32 per 16-batch tile; gates via V_WMMA_F32_16X16X4_F32 with each
// gate (i/f/g/o) mapped to its own 16x16 N-tile so the cell update is
// lane-aligned. Weights live in VGPRs as prebuilt WMMA B-operands.
// Activations use gfx1250 v_tanh_f32 when available.

typedef __attribute__((ext_vector_type(2))) float v2f;
typedef __attribute__((ext_vector_type(8))) float v8f;

namespace {

constexpr int S_LEN = 256;
constexpr int T_IN  = 13;
constexpr int HID   = 10;
constexpr int OUTD  = 9;

#if __has_builtin(__builtin_amdgcn_tanhf)
__device__ __forceinline__ float ftanh(float x) {
  return __builtin_amdgcn_tanhf(x);            // v_tanh_f32 (gfx1250)
}
__device__ __forceinline__ float fsig(float x) {
  // sigmoid(x) = 0.5*tanh(x/2) + 0.5
  return __builtin_fmaf(0.5f, __builtin_amdgcn_tanhf(0.5f * x), 0.5f);
}
#else
__device__ __forceinline__ float fexp(float x) {
  return __builtin_amdgcn_exp2f(x * 1.44269504088896340736f);
}
__device__ __forceinline__ float ftanh(float x) {
  return 1.0f - 2.0f * __builtin_amdgcn_rcpf(fexp(2.0f * x) + 1.0f);
}
__device__ __forceinline__ float fsig(float x) {
  return __builtin_amdgcn_rcpf(1.0f + fexp(-x));
}
#endif

__device__ __forceinline__ v8f wmma4(v2f a, v2f b, v8f c) {
  // D = A(16x4 f32) * B(4x16 f32) + C(16x16 f32)
  return __builtin_amdgcn_wmma_f32_16x16x4_f32(false, a, false, b,
                                               (short)0, c, false, false);
}

} // namespace

__global__ __launch_bounds__(32)
void lstm2_wmma(const float* __restrict__ x,
                const float* __restrict__ W_ih0, const float* __restrict__ W_hh0,
                const float* __restrict__ b_ih0, const float* __restrict__ b_hh0,
                const float* __restrict__ W_ih1, const float* __restrict__ W_hh1,
                const float* __restrict__ b_ih1, const float* __restrict__ b_hh1,
                const float* __restrict__ W_lin, const float* __restrict__ b_lin,
                float* __restrict__ out)
{
  const int lane = threadIdx.x & 31;
  const int half = lane >> 4;   // K-half for A/B operands
  const int col  = lane & 15;   // N column (B/C/D) or M row (A)
  const int b0   = blockIdx.x << 4;

  __shared__ float hstage[16 * 17];  // padded stride vs. 64 banks

  // ---- Build WMMA B-operands (KxN=Kx16 f32 tiles) in registers, once ----
  // Gate g occupies its own N-tile, cols 0..9 valid (cols>=10 are zero).
  v2f Bih0[4][4], Bhh0[4][3], Bih1[4][3], Bhh1[4][3], Blin[3];
  float bias0[4], bias1[4];

  #pragma unroll
  for (int g = 0; g < 4; ++g) {
    const bool rv = (col < HID);
    const int row = g * HID + col;  // row of W (gate-major, torch order i,f,g,o)
    #pragma unroll
    for (int ks = 0; ks < 4; ++ks) {
      const int k0 = 4 * ks + 2 * half;
      v2f r; r.x = 0.0f; r.y = 0.0f;
      if (rv && k0     < T_IN) r.x = W_ih0[row * T_IN + k0];
      if (rv && k0 + 1 < T_IN) r.y = W_ih0[row * T_IN + k0 + 1];
      Bih0[g][ks] = r;
    }
    #pragma unroll
    for (int ks = 0; ks < 3; ++ks) {
      const int k0 = 4 * ks + 2 * half;
      v2f a; a.x = 0.0f; a.y = 0.0f;
      v2f b = a, c = a;
      if (rv && k0 < HID) {
        a.x = W_hh0[row * HID + k0];
        b.x = W_ih1[row * HID + k0];
        c.x = W_hh1[row * HID + k0];
      }
      if (rv && k0 + 1 < HID) {
        a.y = W_hh0[row * HID + k0 + 1];
        b.y = W_ih1[row * HID + k0 + 1];
        c.y = W_hh1[row * HID + k0 + 1];
      }
      Bhh0[g][ks] = a; Bih1[g][ks] = b; Bhh1[g][ks] = c;
    }
    bias0[g] = rv ? (b_ih0[row] + b_hh0[row]) : 0.0f;
    bias1[g] = rv ? (b_ih1[row] + b_hh1[row]) : 0.0f;
  }
  #pragma unroll
  for (int ks = 0; ks < 3; ++ks) {
    const int k0 = 4 * ks + 2 * half;
    v2f r; r.x = 0.0f; r.y = 0.0f;
    if (col < OUTD && k0     < HID) r.x = W_lin[col * HID + k0];
    if (col < OUTD && k0 + 1 < HID) r.y = W_lin[col * HID + k0 + 1];
    Blin[ks] = r;
  }
  const float biasl = (col < OUTD) ? b_lin[col] : 0.0f;

  // ---- Recurrent state: c in D-tile layout, h as prebuilt A-operands ----
  v8f c0, c1;
  v2f ah0[3], ah1[3];
  #pragma unroll
  for (int r = 0; r < 8; ++r) { c0[r] = 0.0f; c1[r] = 0.0f; }
  #pragma unroll
  for (int ks = 0; ks < 3; ++ks) {
    ah0[ks].x = 0.0f; ah0[ks].y = 0.0f;
    ah1[ks].x = 0.0f; ah1[ks].y = 0.0f;
  }

  const float* xr   = x + ((size_t)(b0 + col) * S_LEN) * T_IN;
  float*       outp = out + (size_t)(b0 + 8 * half) * (S_LEN * OUTD) + col;

  #pragma unroll 1
  for (int t = 0; t < S_LEN; ++t) {
    // ---- x_t A-operand: 16 batch rows x K=16 (zero-padded past 13) ----
    // ks=0..2 cover k<=11 (< 13) for both halves: unconditional loads.
    v2f ax[4];
    #pragma unroll
    for (int ks = 0; ks < 3; ++ks) {
      const int k0 = 4 * ks + 2 * half;
      ax[ks].x = xr[k0];
      ax[ks].y = xr[k0 + 1];
    }
    {
      const float x12 = xr[12];          // valid address for all lanes
      ax[3].x = half ? 0.0f : x12;       // only k=12 (half 0) is real
      ax[3].y = 0.0f;
    }
    __builtin_prefetch(xr + T_IN, 0, 1); // next timestep (contiguous)

    // ---- layer 0 gates ----
    v8f gt[4];
    #pragma unroll
    for (int g = 0; g < 4; ++g) {
      v8f d;
      #pragma unroll
      for (int r = 0; r < 8; ++r) d[r] = bias0[g];
      #pragma unroll
      for (int ks = 0; ks < 4; ++ks) d = wmma4(ax[ks],  Bih0[g][ks], d);
      #pragma unroll
      for (int ks = 0; ks < 3; ++ks) d = wmma4(ah0[ks], Bhh0[g][ks], d);
      gt[g] = d;
    }
    v8f h0;
    #pragma unroll
    for (int r = 0; r < 8; ++r) {
      const float ig = fsig(gt[0][r]);
      const float fg = fsig(gt[1][r]);
      const float gg = ftanh(gt[2][r]);
      const float og = fsig(gt[3][r]);
      const float cn = fg * c0[r] + ig * gg;
      c0[r] = cn;
      h0[r] = og * ftanh(cn);
    }
    // D-layout -> A-operand transpose through LDS
    #pragma unroll
    for (int r = 0; r < 8; ++r) hstage[(r + 8 * half) * 17 + col] = h0[r];
    __syncthreads();
    #pragma unroll
    for (int ks = 0; ks < 2; ++ks) {
      const int k0 = 4 * ks + 2 * half;
      ah0[ks].x = hstage[col * 17 + k0];
      ah0[ks].y = hstage[col * 17 + k0 + 1];
    }
    {
      const float h8 = hstage[col * 17 + 8];
      const float h9 = hstage[col * 17 + 9];
      ah0[2].x = half ? 0.0f : h8;       // k=10,11 are padding for half 1
      ah0[2].y = half ? 0.0f : h9;
    }
    __syncthreads();

    // ---- layer 1 gates (ah0 = new h0, ah1 = previous h1) ----
    #pragma unroll
    for (int g = 0; g < 4; ++g) {
      v8f d;
      #pragma unroll
      for (int r = 0; r < 8; ++r) d[r] = bias1[g];
      #pragma unroll
      for (int ks = 0; ks < 3; ++ks) d = wmma4(ah0[ks], Bih1[g][ks], d);
      #pragma unroll
      for (int ks = 0; ks < 3; ++ks) d = wmma4(ah1[ks], Bhh1[g][ks], d);
      gt[g] = d;
    }
    v8f h1;
    #pragma unroll
    for (int r = 0; r < 8; ++r) {
      const float ig = fsig(gt[0][r]);
      const float fg = fsig(gt[1][r]);
      const float gg = ftanh(gt[2][r]);
      const float og = fsig(gt[3][r]);
      const float cn = fg * c1[r] + ig * gg;
      c1[r] = cn;
      h1[r] = og * ftanh(cn);
    }
    #pragma unroll
    for (int r = 0; r < 8; ++r) hstage[(r + 8 * half) * 17 + col] = h1[r];
    __syncthreads();
    #pragma unroll
    for (int ks = 0; ks < 2; ++ks) {
      const int k0 = 4 * ks + 2 * half;
      ah1[ks].x = hstage[col * 17 + k0];
      ah1[ks].y = hstage[col * 17 + k0 + 1];
    }
    {
      const float h8 = hstage[col * 17 + 8];
      const float h9 = hstage[col * 17 + 9];
      ah1[2].x = half ? 0.0f : h8;
      ah1[2].y = half ? 0.0f : h9;
    }
    __syncthreads();

    // ---- output projection: out_t = h1 @ W_lin^T + b_lin ----
    v8f od;
    #pragma unroll
    for (int r = 0; r < 8; ++r) od[r] = biasl;
    #pragma unroll
    for (int ks = 0; ks < 3; ++ks) od = wmma4(ah1[ks], Blin[ks], od);

    if (col < OUTD) {
      #pragma unroll
      for (int r = 0; r < 8; ++r)
        __builtin_nontemporal_store(od[r], outp + (size_t)r * (S_LEN * OUTD));
    }
    outp += OUTD;
    xr   += T_IN;
  }
}

extern "C" void kernel_launch(void* const* d_in, const int* in_sizes, int n_in,
                              void* d_out, int out_size, void* d_ws, size_t ws_size,
                              hipStream_t stream) {
  (void)n_in; (void)out_size; (void)d_ws; (void)ws_size;
  const float* x     = (const float*)d_in[0];
  const float* W_ih0 = (const float*)d_in[1];
  const float* W_hh0 = (const float*)d_in[2];
  const float* b_ih0 = (const float*)d_in[3];
  const float* b_hh0 = (const float*)d_in[4];
  const float* W_ih1 = (const float*)d_in[5];
  const float* W_hh1 = (const float*)d_in[6];
  const float* b_ih1 = (const float*)d_in[7];
  const float* b_hh1 = (const float*)d_in[8];
  const float* W_lin = (const float*)d_in[9];
  const float* b_lin = (const float*)d_in[10];
  float* out = (float*)d_out;

  const int Bdim   = in_sizes[0] / (256 * 13);  // 8192
  const int blocks = Bdim / 16;                 // 512 independent wave32 tiles

  lstm2_wmma<<<blocks, 32, 0, stream>>>(x, W_ih0, W_hh0, b_ih0, b_hh0,
                                        W_ih1, W_hh1, b_ih1, b_hh1,
                                        W_lin, b_lin, out);
}